// MainEffectNN_72636486909961
// MI455X (gfx1250) — compile-verified
//
#include <hip/hip_runtime.h>

// MI455X / gfx1250, wave32. fp32 WMMA path: V_WMMA_F32_16X16X4_F32.
// Memory-bound: x = 537 MB streamed once @ 23.3 TB/s => ~23 us floor.

typedef float v2f __attribute__((ext_vector_type(2)));
typedef float v8f __attribute__((ext_vector_type(8)));

#define B_TOT 8192
#define G_TOT 256
#define DMAX  64
#define H_DIM 10
#define GSPLIT 8                 // gene-dimension split for parallelism
#define G_CHUNK (G_TOT / GSPLIT) // 32 genes per block

__global__ __launch_bounds__(256) void init_out_kernel(float* __restrict__ out,
                                                       const float* __restrict__ bh) {
    int i = blockIdx.x * blockDim.x + threadIdx.x;
    if (i < B_TOT) out[i] = bh[0];
}

__global__ __launch_bounds__(256) void gene_mlp_wmma_kernel(
    const float* __restrict__ x,   // [B, G, DMAX]
    const float* __restrict__ W1,  // [G, DMAX, H]
    const float* __restrict__ b1,  // [G, H]
    const float* __restrict__ W2,  // [G, H]
    const float* __restrict__ b2,  // [G]
    const float* __restrict__ Wh,  // [G]
    float* __restrict__ out)       // [B]
{
    // Double-buffered LDS tile: padded W1[g]^T as sB[buf][n][d]. 2 x 4 KB.
    __shared__ float sB[2][16 * 64];

    const int lane = threadIdx.x & 31;
    const int wave = threadIdx.x >> 5;   // 8 waves / block, 16 b-rows each
    const int n    = lane & 15;          // N index (hidden unit / output column)
    const int hi   = lane >> 4;          // half-wave select
    const int koff = hi * 2;             // K sub-offset within each 16x16x4 step
    const int b0   = blockIdx.x * 128 + wave * 16;
    const int g0   = blockIdx.y * G_CHUNK;

    // A-row base for this lane: row m = b0 + n of x
    const float* xrow = x + (size_t)(b0 + n) * (G_TOT * DMAX);

    float acc[8];
#pragma unroll
    for (int v = 0; v < 8; ++v) acc[v] = 0.f;

    // ---- preamble: stage gene g0 into buffer 0 ----
    {
        const float* W1g = W1 + g0 * (DMAX * H_DIM);
#pragma unroll
        for (int i = 0; i < 4; ++i) {
            int e  = (int)threadIdx.x * 4 + i;   // 0..1023
            int nn = e >> 6;
            int dd = e & 63;
            sB[0][e] = (nn < H_DIM) ? W1g[dd * H_DIM + nn] : 0.f;
        }
    }
    __syncthreads();

    for (int gi = 0; gi < G_CHUNK; ++gi) {
        const int g   = g0 + gi;
        const int cur = gi & 1;

        // ---- stage NEXT gene into the other buffer (overlaps with compute) ----
        if (gi + 1 < G_CHUNK) {
            const float* W1n = W1 + (g + 1) * (DMAX * H_DIM);
#pragma unroll
            for (int i = 0; i < 4; ++i) {
                int e  = (int)threadIdx.x * 4 + i;
                int nn = e >> 6;
                int dd = e & 63;
                sB[cur ^ 1][e] = (nn < H_DIM) ? W1n[dd * H_DIM + nn] : 0.f;
            }
        }

        const float b1n = (n < H_DIM) ? b1[g * H_DIM + n] : 0.f;
        const float w2n = (n < H_DIM) ? W2[g * H_DIM + n] : 0.f;
        const float whg = Wh[g];
        const float cgb = (n == 0) ? whg * b2[g] : 0.f;  // b2 term, once per row m

        const float* ap = xrow + g * DMAX + koff;        // x[b0+n][g][koff + 4k ..]
        const float* bp = &sB[cur][n * 64 + koff];       // W1^T[n][koff + 4k ..]

        // ---- prefetch ALL 16 A-fragments first: max loads in flight ----
        v2f afrag[16];
#pragma unroll
        for (int k = 0; k < 16; ++k)
            afrag[k] = *(const v2f*)(ap + 4 * k);

        // ---- D = x_tile(16x64) @ W1pad(64x16), K blocked by 4 ----
        v8f c = {0.f, 0.f, 0.f, 0.f, 0.f, 0.f, 0.f, 0.f};
#pragma unroll
        for (int k = 0; k < 16; ++k) {
            v2f b = *(const v2f*)(bp + 4 * k);
            c = __builtin_amdgcn_wmma_f32_16x16x4_f32(
                    /*neg_a=*/false, afrag[k], /*neg_b=*/false, b,
                    /*c_mod=*/(short)0, c, /*reuse_a=*/false, /*reuse_b=*/false);
        }

        // ---- linear epilogue: fold W2, Wh, b1, b2 per lane ----
        const float coef = whg * w2n;
#pragma unroll
        for (int v = 0; v < 8; ++v) {
            float hv = fmaxf(c[v] + b1n, 0.f);
            acc[v] += coef * hv + cgb;
        }

        // one barrier per gene: next-buffer stores done, current-buffer reads done
        __syncthreads();
    }

    // Reduce over the 16 N-lanes of each half-wave; M = v + 8*hi per C/D layout.
#pragma unroll
    for (int v = 0; v < 8; ++v) {
        float s = acc[v];
        s += __shfl_xor(s, 1, 32);
        s += __shfl_xor(s, 2, 32);
        s += __shfl_xor(s, 4, 32);
        s += __shfl_xor(s, 8, 32);
        if (n == 0) atomicAdd(&out[b0 + hi * 8 + v], s);
    }
}

extern "C" void kernel_launch(void* const* d_in, const int* in_sizes, int n_in,
                              void* d_out, int out_size, void* d_ws, size_t ws_size,
                              hipStream_t stream) {
    const float* x  = (const float*)d_in[0];
    const float* W1 = (const float*)d_in[1];
    const float* b1 = (const float*)d_in[2];
    const float* W2 = (const float*)d_in[3];
    const float* b2 = (const float*)d_in[4];
    const float* Wh = (const float*)d_in[5];
    const float* bh = (const float*)d_in[6];
    float* out = (float*)d_out;

    // out[b] = bh (harness poisons d_out; re-init every launch)
    init_out_kernel<<<dim3((B_TOT + 255) / 256), dim3(256), 0, stream>>>(out, bh);

    // 64 b-blocks x 8 gene-chunks = 512 blocks, 8 waves each
    gene_mlp_wmma_kernel<<<dim3(B_TOT / 128, GSPLIT), dim3(256), 0, stream>>>(
        x, W1, b1, W2, b2, Wh, out);
}